// NCEAverage_66967130079741
// MI455X (gfx1250) — compile-verified
//
#include <hip/hip_runtime.h>
#include <hip/hip_bf16.h>

// NCE forward for MI455X (gfx1250).
//
// Shapes: x[256,128] f32, y[256] i32, idx[256,4097] i32, memory[1280000,128] f32.
// out[b,k] = exp(dot(memory[row(b,k)], x[b]) / T) / Z,  row(b,0)=y[b], else idx[b,k].
// Z = mean(e) * 1280000 (global constant).
//
// Memory-bound: ~537 MB of gathered f32 rows vs 268 MFLOP of dot products.
// Dots are done in full f32 precision on the matrix pipe via
// V_WMMA_F32_16X16X4_F32 (A = 16 gathered rows x 4-chunk of D, B = x[b] chunk
// broadcast across all 16 columns -> all output columns equal the wanted dots).

#define D_DIM      128
#define KP1        4097           // K+1
#define NBATCH     256
#define OUTPUTSZ   1280000
#define INV_T      (1.0f / 0.07f)
#define KTILES_PER_BLOCK 8        // 8 waves, 16 k-rows each -> 128 k per block
#define BLOCKS_PER_BATCH 33       // ceil(4097 / 128)
#define NPARTIALS  (BLOCKS_PER_BATCH * NBATCH)

typedef __attribute__((ext_vector_type(2))) float v2f;
typedef __attribute__((ext_vector_type(8))) float v8f;

__global__ __launch_bounds__(256)
void nce_fwd_kernel(const float* __restrict__ x,
                    const int*   __restrict__ y,
                    const int*   __restrict__ idx,
                    const float* __restrict__ mem,
                    float*       __restrict__ e_out,
                    float*       __restrict__ partials)
{
    const int b    = blockIdx.y;
    const int wave = threadIdx.x >> 5;          // 0..7
    const int lane = threadIdx.x & 31;
    const int half = lane >> 4;                 // 0: K=0,1 | 1: K=2,3 (A layout)
    const int mrow = lane & 15;                 // A-matrix row this lane feeds
    const int sel  = half * 2;                  // float offset inside 4-chunk
    const int kbase = blockIdx.x * (KTILES_PER_BLOCK * 16) + wave * 16;

    // Row index for this lane's A row (both half-groups load the same row set).
    const int kk = kbase + mrow;
    int row = 0;
    if (kk < KP1)
        row = (kk == 0) ? y[b] : idx[(size_t)b * KP1 + kk];

    const float* __restrict__ arow = mem + (size_t)row * D_DIM + sel;
    const float* __restrict__ xrow = x   + (size_t)b   * D_DIM + sel;

    // 32 chained f32 WMMAs, K=4 each, cover D=128.
    v8f acc = {0.f, 0.f, 0.f, 0.f, 0.f, 0.f, 0.f, 0.f};
#pragma unroll
    for (int c = 0; c < 32; ++c) {
        v2f a;  a.x  = arow[c * 4]; a.y  = arow[c * 4 + 1];   // gathered row chunk
        v2f bb; bb.x = xrow[c * 4]; bb.y = xrow[c * 4 + 1];   // x chunk, all cols equal
        acc = __builtin_amdgcn_wmma_f32_16x16x4_f32(
            /*neg_a=*/false, a, /*neg_b=*/false, bb,
            /*c_mod=*/(short)0, acc, /*reuse_a=*/false, /*reuse_b=*/false);
    }

    // C layout: VGPR r -> M=r (lanes 0-15) / M=8+r (lanes 16-31); columns identical.
    float ev[8];
    float lsum = 0.0f;
#pragma unroll
    for (int r = 0; r < 8; ++r) {
        const int m = r + half * 8;
        const int kcur = kbase + m;
        float e = 0.0f;
        if (kcur < KP1) e = __expf(acc[r] * INV_T);
        ev[r] = e;
        lsum += e;
    }

    // Column 0 (lanes 0 and 16) writes the 16 e-values of this tile.
    if (mrow == 0) {
#pragma unroll
        for (int r = 0; r < 8; ++r) {
            const int kcur = kbase + r + half * 8;
            if (kcur < KP1) e_out[(size_t)b * KP1 + kcur] = ev[r];
        }
    }

    // Deterministic wave reduction. All 16 columns duplicate the sum -> /16 exact.
#pragma unroll
    for (int off = 16; off > 0; off >>= 1)
        lsum += __shfl_down(lsum, off, 32);

    __shared__ float wsum[KTILES_PER_BLOCK];
    if (lane == 0) wsum[wave] = lsum * (1.0f / 16.0f);
    __syncthreads();
    if (threadIdx.x == 0) {
        float s = 0.0f;
        for (int i = 0; i < KTILES_PER_BLOCK; ++i) s += wsum[i];
        partials[b * BLOCKS_PER_BATCH + blockIdx.x] = s;
    }
}

// Fixed-order reduction of block partials -> scale = 1/Z.
__global__ __launch_bounds__(256)
void nce_reduce_kernel(const float* __restrict__ partials,
                       float* __restrict__ scale)
{
    __shared__ float sh[256];
    float s = 0.0f;
    for (int i = threadIdx.x; i < NPARTIALS; i += 256) s += partials[i];
    sh[threadIdx.x] = s;
    __syncthreads();
    for (int off = 128; off > 0; off >>= 1) {
        if (threadIdx.x < off) sh[threadIdx.x] += sh[threadIdx.x + off];
        __syncthreads();
    }
    if (threadIdx.x == 0) {
        const float mean = sh[0] / (float)((size_t)NBATCH * KP1);
        const float Z = mean * (float)OUTPUTSZ;
        *scale = 1.0f / Z;
    }
}

__global__ __launch_bounds__(256)
void nce_scale_kernel(float* __restrict__ out, const float* __restrict__ scale, int n)
{
    const int i = blockIdx.x * blockDim.x + threadIdx.x;
    const float s = *scale;
    if (i < n) out[i] *= s;
}

extern "C" void kernel_launch(void* const* d_in, const int* in_sizes, int n_in,
                              void* d_out, int out_size, void* d_ws, size_t ws_size,
                              hipStream_t stream)
{
    const float* x   = (const float*)d_in[0];
    const int*   y   = (const int*)  d_in[1];
    const int*   idx = (const int*)  d_in[2];
    const float* mem = (const float*)d_in[3];
    float* out = (float*)d_out;

    float* partials = (float*)d_ws;               // NPARTIALS floats
    float* scale    = partials + NPARTIALS;       // 1 float

    dim3 grid(BLOCKS_PER_BATCH, NBATCH);
    nce_fwd_kernel<<<grid, 256, 0, stream>>>(x, y, idx, mem, out, partials);
    nce_reduce_kernel<<<1, 256, 0, stream>>>(partials, scale);

    const int n = NBATCH * KP1;
    nce_scale_kernel<<<(n + 255) / 256, 256, 0, stream>>>(out, scale, n);
}